// MakeCutouts_31774168056217
// MI455X (gfx1250) — compile-verified
//
#include <hip/hip_runtime.h>
#include <stdint.h>

#define CUT_SIZE 224
#define CUTN     128
#define SIDE     1024
#define HW_IN    (SIDE * SIDE)
#define HW_OUT   (CUT_SIZE * CUT_SIZE)

// One block per (cut, output row). Stage input rows y0 & y1 (3 channels) into
// LDS via CDNA5 async global->LDS DMA (b128), then bilinear-interpolate from
// LDS with non-temporal coalesced stores.
__global__ __launch_bounds__(256) void make_cutouts_kernel(
    const float* __restrict__ img,   // [3, 1024, 1024]
    const int*   __restrict__ sizes, // [128]
    const int*   __restrict__ offy,  // [128]
    const int*   __restrict__ offx,  // [128]
    float*       __restrict__ out)   // [128, 3, 224, 224]
{
    __shared__ __align__(16) float smem[2][3][1024]; // [y0/y1][channel][staged column]

    const int blk = blockIdx.x;
    const int n   = blk / CUT_SIZE;   // cut index
    const int i   = blk % CUT_SIZE;   // output row
    const int tid = threadIdx.x;

    const float fsize = (float)sizes[n];
    const float oy    = (float)offy[n];
    const float ox    = (float)offx[n];
    const float scale = fsize * (1.0f / (float)CUT_SIZE);

    // --- row coordinate (uniform across block) ---
    float ysf = oy + ((float)i + 0.5f) * scale - 0.5f;
    ysf = fminf(fmaxf(ysf, 0.0f), (float)(SIDE - 1));
    const int   y0 = (int)floorf(ysf);
    const int   y1 = (y0 + 1 < SIDE - 1) ? (y0 + 1) : (SIDE - 1);
    const float wy = ysf - (float)y0;

    // --- needed column window [xbeg, xend] (uniform) ---
    float xsA = ox + 0.5f * scale - 0.5f;
    xsA = fminf(fmaxf(xsA, 0.0f), (float)(SIDE - 1));
    float xsB = ox + ((float)CUT_SIZE - 0.5f) * scale - 0.5f;
    xsB = fminf(fmaxf(xsB, 0.0f), (float)(SIDE - 1));
    const int xbeg  = (int)floorf(xsA);
    int xend        = (int)floorf(xsB) + 1;
    if (xend > SIDE - 1) xend = SIDE - 1;

    // Vectorized staging window: start at a 4-column boundary so each b128
    // copy is 16B-aligned. Last vector start <= 1020 (multiple of 4, <= xend),
    // so no vector crosses the row end -> no OOB reads, slots < 1024.
    const int xbeg4 = xbeg & ~3;
    const int nvec  = ((xend - xbeg4) >> 2) + 1;   // <= 256 b128 copies per row

    // --- phase 1: async DMA rows y0/y1 of all 3 channels into LDS ---
    const uint32_t lds_base = (uint32_t)(uintptr_t)&smem[0][0][0]; // LDS byte offset
    const uint64_t gbase    = (uint64_t)(uintptr_t)img;

    for (int v = tid; v < nvec; v += 256) {
        const int col  = xbeg4 + 4 * v;   // global start column (16B aligned)
        const int slot = 4 * v;           // LDS slot (16B aligned)
#pragma unroll
        for (int c = 0; c < 3; ++c) {
            const uint32_t l0 = lds_base + (uint32_t)(((0 * 3 + c) * 1024 + slot) * 4);
            const uint32_t l1 = lds_base + (uint32_t)(((1 * 3 + c) * 1024 + slot) * 4);
            const uint64_t g0 = gbase + (uint64_t)(uint32_t)(((c * SIDE + y0) * SIDE + col) * 4);
            const uint64_t g1 = gbase + (uint64_t)(uint32_t)(((c * SIDE + y1) * SIDE + col) * 4);
            asm volatile("global_load_async_to_lds_b128 %0, %1, off"
                         :: "v"(l0), "v"(g0) : "memory");
            asm volatile("global_load_async_to_lds_b128 %0, %1, off"
                         :: "v"(l1), "v"(g1) : "memory");
        }
    }
    asm volatile("s_wait_asynccnt 0x0" ::: "memory");
    __syncthreads();

    // --- phase 2: bilinear from LDS, clamp, NT coalesced store ---
    const float omwy = 1.0f - wy;
    float* outn = out + (size_t)n * 3 * HW_OUT + (size_t)i * CUT_SIZE;

    for (int idx = tid; idx < 3 * CUT_SIZE; idx += 256) {
        const int c = idx / CUT_SIZE;
        const int j = idx - c * CUT_SIZE;

        float xsf = ox + ((float)j + 0.5f) * scale - 0.5f;
        xsf = fminf(fmaxf(xsf, 0.0f), (float)(SIDE - 1));
        int x0 = (int)floorf(xsf);
        int x1 = (x0 + 1 < SIDE - 1) ? (x0 + 1) : (SIDE - 1);
        const float wx = xsf - (float)x0;
        x0 -= xbeg4;
        x1 -= xbeg4;

        const float r0 = smem[0][c][x0] * omwy + smem[1][c][x0] * wy;
        const float r1 = smem[0][c][x1] * omwy + smem[1][c][x1] * wy;
        float v = r0 * (1.0f - wx) + r1 * wx;
        v = fminf(fmaxf(v, 0.0f), 1.0f);   // ClampWithGrad forward

        __builtin_nontemporal_store(v, &outn[(size_t)c * HW_OUT + j]);
    }
}

extern "C" void kernel_launch(void* const* d_in, const int* in_sizes, int n_in,
                              void* d_out, int out_size, void* d_ws, size_t ws_size,
                              hipStream_t stream) {
    (void)in_sizes; (void)n_in; (void)out_size; (void)d_ws; (void)ws_size;
    const float* img   = (const float*)d_in[0];
    const int*   sizes = (const int*)d_in[1];
    const int*   oy    = (const int*)d_in[2];
    const int*   ox    = (const int*)d_in[3];
    float*       out   = (float*)d_out;

    dim3 grid(CUTN * CUT_SIZE);   // 28672 blocks: one per (cut, output row)
    dim3 block(256);              // 8 wave32 waves
    make_cutouts_kernel<<<grid, block, 0, stream>>>(img, sizes, oy, ox, out);
}